// GCN_VCG_14104672600357
// MI455X (gfx1250) — compile-verified
//
#include <hip/hip_runtime.h>
#include <hip/hip_bf16.h>

typedef __attribute__((ext_vector_type(2))) float v2f;
typedef __attribute__((ext_vector_type(4))) float v4f;
typedef __attribute__((ext_vector_type(8))) float v8f;

// Explicit global-address-space pointer types: keep A/W/Y traffic on the
// GLOBAL_* path (LOADcnt/STOREcnt only). FLAT ops would also consume DScnt and
// contend with the LDS pipe feeding the WMMA B-fragments.
// NOTE: use trivial ext-vector types (not HIP_vector_type) through AS(1).
typedef const float __attribute__((address_space(1)))  gcfloat;
typedef float       __attribute__((address_space(1)))  gfloat;
typedef const v4f   __attribute__((address_space(1)))  gcv4f;
typedef const int   __attribute__((address_space(1)))  gcint;

// ---------------------------------------------------------------------------
// Generic WMMA fp32 GEMM:  Y[M,128] = act( concat(X0,X1,X2)[M, nblk*128] @ W + b )
// W is row-major [nblk*128, 128]. 256 threads = 8 wave32s, each wave computes a
// 16x128 output tile via v_wmma_f32_16x16x4_f32 (K stepped by 4).
// ---------------------------------------------------------------------------
__global__ __launch_bounds__(256) void gemm128_wmma(
    const float* __restrict__ X0, const float* __restrict__ X1,
    const float* __restrict__ X2, int nblk,
    const float* __restrict__ W, const float* __restrict__ bias,
    float* __restrict__ Y, int M, int relu)
{
  __shared__ float ldsW[128 * 128];   // one 128x128 K-block of W, xor-swizzled

  const int tid  = threadIdx.x;
  const int wave = tid >> 5;          // 0..7 (wave32)
  const int lane = tid & 31;
  const int rowBase = blockIdx.x * 128 + wave * 16;

  const int m    = lane & 15;         // matrix row (A) / column (B,C,D) in-tile
  const int half = lane >> 4;         // lane half selects K pair / M half
  const int kk   = half * 2;          // K offset within the 4-wide step
  const int swz  = half << 4;         // LDS bank-conflict xor swizzle

  v8f acc[8] = {};                    // 16x128 fp32 accumulators (8 N-tiles)

  int rowA = rowBase + m;
  if (rowA >= M) rowA = M - 1;        // clamp; OOB rows masked at store

  for (int blk = 0; blk < nblk; ++blk) {
    __syncthreads();
    // Stage W K-block into LDS, coalesced 128-bit loads, xor-16 column swizzle
    // on (k>>1)&1 so lanes 0-15 (read row k) and 16-31 (read row k+2) use
    // disjoint bank groups.
    {
      gcv4f* Wv = (gcv4f*)(W + (size_t)blk * 128 * 128);
      for (int i = tid; i < 128 * 32; i += 256) {
        int k  = i >> 5;
        int c4 = (i & 31) << 2;
        int sc = c4 ^ (((k >> 1) & 1) << 4);
        v4f t = Wv[i];
        *(v4f*)(&ldsW[k * 128 + sc]) = t;
      }
    }
    __syncthreads();

    const float* Xsel = (blk == 0) ? X0 : ((blk == 1) ? X1 : X2);
    gcfloat* xrow = (gcfloat*)(Xsel + (size_t)rowA * 128);
#pragma unroll 4
    for (int k0 = 0; k0 < 128; k0 += 4) {
      v2f a;
      a.x = xrow[k0 + kk];
      a.y = xrow[k0 + kk + 1];
#pragma unroll
      for (int nt = 0; nt < 8; ++nt) {
        int c = (nt * 16 + m) ^ swz;  // swizzled LDS column
        v2f b;
        b.x = ldsW[(k0 + kk) * 128 + c];
        b.y = ldsW[(k0 + kk + 1) * 128 + c];
        acc[nt] = __builtin_amdgcn_wmma_f32_16x16x4_f32(
            false, a, false, b, (short)0, acc[nt], false, false);
      }
    }
  }

  // Epilogue: C/D layout -> VGPR r holds M = r + 8*half, N = nt*16 + m.
  gcfloat* gb = (gcfloat*)bias;
  gfloat*  gy = (gfloat*)Y;
#pragma unroll
  for (int nt = 0; nt < 8; ++nt) {
    float bb = gb[nt * 16 + m];
#pragma unroll
    for (int r = 0; r < 8; ++r) {
      int row = rowBase + half * 8 + r;
      if (row < M) {
        float v = acc[nt][r] + bb;
        if (relu) v = fmaxf(v, 0.0f);
        gy[(size_t)row * 128 + nt * 16 + m] = v;
      }
    }
  }
}

// ---------------------------------------------------------------------------
// Edge preprocessing: endpoint gather + degree histograms.
// ---------------------------------------------------------------------------
__global__ void build_edges(const int* __restrict__ v_ei, const int* __restrict__ c_ei,
                            const int* __restrict__ p_ei, const int* __restrict__ n_ei,
                            int* __restrict__ p_v, int* __restrict__ p_c,
                            int* __restrict__ n_v, int* __restrict__ n_c,
                            float* __restrict__ degPv, float* __restrict__ degNv,
                            float* __restrict__ degPc, float* __restrict__ degNc, int Eh)
{
  int e = blockIdx.x * blockDim.x + threadIdx.x;
  if (e >= Eh) return;
  int i = p_ei[e];
  int pv = v_ei[i], pc = c_ei[i];
  p_v[e] = pv; p_c[e] = pc;
  __hip_atomic_fetch_add((gfloat*)(degPv + pv), 1.0f, __ATOMIC_RELAXED, __HIP_MEMORY_SCOPE_AGENT);
  __hip_atomic_fetch_add((gfloat*)(degPc + pc), 1.0f, __ATOMIC_RELAXED, __HIP_MEMORY_SCOPE_AGENT);
  i = n_ei[e];
  int nv = v_ei[i], nc = c_ei[i];
  n_v[e] = nv; n_c[e] = nc;
  __hip_atomic_fetch_add((gfloat*)(degNv + nv), 1.0f, __ATOMIC_RELAXED, __HIP_MEMORY_SCOPE_AGENT);
  __hip_atomic_fetch_add((gfloat*)(degNc + nc), 1.0f, __ATOMIC_RELAXED, __HIP_MEMORY_SCOPE_AGENT);
}

__global__ void compute_rnorm(const int* __restrict__ ev, const int* __restrict__ ec,
                              const float* __restrict__ degV, const float* __restrict__ degC,
                              float* __restrict__ rnorm, int Eh)
{
  int e = blockIdx.x * blockDim.x + threadIdx.x;
  if (e >= Eh) return;
  float dv = fmaxf(((gcfloat*)degV)[ev[e]], 1.0f);
  float dc = fmaxf(((gcfloat*)degC)[ec[e]], 1.0f);
  rnorm[e] = 1.0f / sqrtf(dv * dc);
}

// ---------------------------------------------------------------------------
// Fused gather * (1/norm) -> atomic scatter-add. One wave32 per edge,
// 128-bit load per lane (128 floats / row). global_atomic_add_f32, relaxed/agent.
// ---------------------------------------------------------------------------
__global__ __launch_bounds__(256) void scatter_edges(
    const float* __restrict__ T, const int* __restrict__ src,
    const int* __restrict__ dst, const float* __restrict__ rnorm,
    float* __restrict__ Agg, int Eh)
{
  int e = blockIdx.x * (blockDim.x >> 5) + (threadIdx.x >> 5);
  if (e >= Eh) return;
  int lane = threadIdx.x & 31;
  float s = ((gcfloat*)rnorm)[e];
  int si = ((gcint*)src)[e];
  int di = ((gcint*)dst)[e];
  v4f v = ((gcv4f*)(T + (size_t)si * 128))[lane];
  gfloat* out = (gfloat*)(Agg + (size_t)di * 128 + lane * 4);
  __hip_atomic_fetch_add(out + 0, v.x * s, __ATOMIC_RELAXED, __HIP_MEMORY_SCOPE_AGENT);
  __hip_atomic_fetch_add(out + 1, v.y * s, __ATOMIC_RELAXED, __HIP_MEMORY_SCOPE_AGENT);
  __hip_atomic_fetch_add(out + 2, v.z * s, __ATOMIC_RELAXED, __HIP_MEMORY_SCOPE_AGENT);
  __hip_atomic_fetch_add(out + 3, v.w * s, __ATOMIC_RELAXED, __HIP_MEMORY_SCOPE_AGENT);
}

// ---------------------------------------------------------------------------
extern "C" void kernel_launch(void* const* d_in, const int* in_sizes, int n_in,
                              void* d_out, int out_size, void* d_ws, size_t ws_size,
                              hipStream_t stream)
{
  const int* v_ei = (const int*)d_in[2];
  const int* c_ei = (const int*)d_in[3];
  const int* p_ei = (const int*)d_in[4];
  const int* n_ei = (const int*)d_in[5];
  const float* v_emb0 = (const float*)d_in[6];
  const float* c_emb0 = (const float*)d_in[7];
  const float* Wmlp[4] = { (const float*)d_in[8],  (const float*)d_in[10],
                           (const float*)d_in[12], (const float*)d_in[14] };
  const float* Bmlp[4] = { (const float*)d_in[9],  (const float*)d_in[11],
                           (const float*)d_in[13], (const float*)d_in[15] };
  const float* cu_W = (const float*)d_in[16];
  const float* cu_b = (const float*)d_in[17];
  const float* vu_W = (const float*)d_in[18];
  const float* vu_b = (const float*)d_in[19];

  const int V  = in_sizes[6] / 128;
  const int C  = in_sizes[7] / 128;
  const int Eh = in_sizes[4];
  const int NIT = 3;
  const size_t dV = (size_t)V * 128, dC = (size_t)C * 128;

  // ---- workspace carve -----------------------------------------------------
  char* w = (char*)d_ws;
  auto take = [&](size_t bytes) -> char* {
    char* p = w; w += (bytes + 255) & ~(size_t)255; return p;
  };
  float* degPv = (float*)take(((size_t)2 * V + 2 * C) * sizeof(float));
  float* degNv = degPv + V;
  float* degPc = degNv + V;
  float* degNc = degPc + C;
  int*   p_v   = (int*)take((size_t)Eh * 4);
  int*   p_c   = (int*)take((size_t)Eh * 4);
  int*   n_v   = (int*)take((size_t)Eh * 4);
  int*   n_c   = (int*)take((size_t)Eh * 4);
  float* p_rn  = (float*)take((size_t)Eh * 4);
  float* n_rn  = (float*)take((size_t)Eh * 4);
  float* h_v   = (float*)take(dV * 4);            // MLP hidden (v side)
  float* h_c   = (float*)take(dC * 4);            // MLP hidden (c side)
  float* m_pv2c = (float*)take(dV * 4);           // MLP outputs
  float* m_nv2c = (float*)take(dV * 4);
  float* m_pc2v = (float*)take(dC * 4);
  float* m_nc2v = (float*)take(dC * 4);
  float* aggs   = (float*)take((2 * dC + 2 * dV) * 4);
  float* agg_pv2c = aggs;                         // [C,128]
  float* agg_nv2c = agg_pv2c + dC;                // [C,128]
  float* agg_pc2v = agg_nv2c + dC;                // [V,128]
  float* agg_nc2v = agg_pc2v + dV;                // [V,128]

  // ---- output slices: stack(v_embs)[4,V,128] then stack(c_embs)[4,C,128] ---
  float* out_v = (float*)d_out;
  float* out_c = out_v + (size_t)(NIT + 1) * dV;

  (void)hipMemcpyAsync(out_v, v_emb0, dV * 4, hipMemcpyDeviceToDevice, stream);
  (void)hipMemcpyAsync(out_c, c_emb0, dC * 4, hipMemcpyDeviceToDevice, stream);

  // ---- degrees + per-edge 1/norm (cheap; recomputed each call) -------------
  (void)hipMemsetAsync(degPv, 0, ((size_t)2 * V + 2 * C) * 4, stream);
  int gE = (Eh + 255) / 256;
  build_edges<<<gE, 256, 0, stream>>>(v_ei, c_ei, p_ei, n_ei,
                                      p_v, p_c, n_v, n_c,
                                      degPv, degNv, degPc, degNc, Eh);
  compute_rnorm<<<gE, 256, 0, stream>>>(p_v, p_c, degPv, degPc, p_rn, Eh);
  compute_rnorm<<<gE, 256, 0, stream>>>(n_v, n_c, degNv, degNc, n_rn, Eh);

  const int gV = (V + 127) / 128;
  const int gC = (C + 127) / 128;
  const int gS = (Eh + 7) / 8;

  for (int it = 0; it < NIT; ++it) {
    const float* v_cur = out_v + (size_t)it * dV;
    const float* c_cur = out_c + (size_t)it * dC;
    float* v_nxt = out_v + (size_t)(it + 1) * dV;
    float* c_nxt = out_c + (size_t)(it + 1) * dC;

    // 2-layer MLPs (relu hidden, linear out). Wmlp[i] = [2,128,128], b = [2,128].
    // pv2c / nv2c on variables:
    gemm128_wmma<<<gV, 256, 0, stream>>>(v_cur, nullptr, nullptr, 1,
        Wmlp[0], Bmlp[0], h_v, V, 1);
    gemm128_wmma<<<gV, 256, 0, stream>>>(h_v, nullptr, nullptr, 1,
        Wmlp[0] + 128 * 128, Bmlp[0] + 128, m_pv2c, V, 0);
    gemm128_wmma<<<gV, 256, 0, stream>>>(v_cur, nullptr, nullptr, 1,
        Wmlp[1], Bmlp[1], h_v, V, 1);
    gemm128_wmma<<<gV, 256, 0, stream>>>(h_v, nullptr, nullptr, 1,
        Wmlp[1] + 128 * 128, Bmlp[1] + 128, m_nv2c, V, 0);
    // pc2v / nc2v on clauses:
    gemm128_wmma<<<gC, 256, 0, stream>>>(c_cur, nullptr, nullptr, 1,
        Wmlp[2], Bmlp[2], h_c, C, 1);
    gemm128_wmma<<<gC, 256, 0, stream>>>(h_c, nullptr, nullptr, 1,
        Wmlp[2] + 128 * 128, Bmlp[2] + 128, m_pc2v, C, 0);
    gemm128_wmma<<<gC, 256, 0, stream>>>(c_cur, nullptr, nullptr, 1,
        Wmlp[3], Bmlp[3], h_c, C, 1);
    gemm128_wmma<<<gC, 256, 0, stream>>>(h_c, nullptr, nullptr, 1,
        Wmlp[3] + 128 * 128, Bmlp[3] + 128, m_nc2v, C, 0);

    // Normalized segment sums along both directions.
    (void)hipMemsetAsync(aggs, 0, (2 * dC + 2 * dV) * 4, stream);
    scatter_edges<<<gS, 256, 0, stream>>>(m_pv2c, p_v, p_c, p_rn, agg_pv2c, Eh);
    scatter_edges<<<gS, 256, 0, stream>>>(m_nv2c, n_v, n_c, n_rn, agg_nv2c, Eh);
    scatter_edges<<<gS, 256, 0, stream>>>(m_pc2v, p_c, p_v, p_rn, agg_pc2v, Eh);
    scatter_edges<<<gS, 256, 0, stream>>>(m_nc2v, n_c, n_v, n_rn, agg_nc2v, Eh);

    // Concat-free update GEMMs: [*,384] @ [384,128] + b.
    gemm128_wmma<<<gC, 256, 0, stream>>>(c_cur, agg_pv2c, agg_nv2c, 3,
        cu_W, cu_b, c_nxt, C, 0);
    gemm128_wmma<<<gV, 256, 0, stream>>>(v_cur, agg_pc2v, agg_nc2v, 3,
        vu_W, vu_b, v_nxt, V, 0);
  }

  (void)n_in; (void)out_size; (void)ws_size;
}